// ConvKernelBank_10307921510566
// MI455X (gfx1250) — compile-verified
//
#include <hip/hip_runtime.h>
#include <hip/hip_bf16.h>
#include <math.h>

// Problem constants (from reference setup_inputs)
static constexpr int B = 4, C = 128, T = 32, H = 24, W = 24;
static constexpr int M = 3, KT = 3, KS = 3, MIXK = 3;
static constexpr int HW = H * W;              // 576
static constexpr int N = B * T;               // 128 (GEMM N dim)
static constexpr int PW = W + 2;              // padded plane width 26
static constexpr int CHUNK = 8;               // time-steps per conv block
static constexpr int NCHUNK = T / CHUNK;      // 4
static constexpr size_t PLANE_ELEMS = (size_t)B * C * T * HW; // 9437184

typedef __attribute__((ext_vector_type(2))) float v2f;
typedef __attribute__((ext_vector_type(8))) float v8f;

#if __has_builtin(__builtin_amdgcn_global_load_async_to_lds_b32)
#define USE_ASYNC_LDS 1
// Builtin signature (from clang diagnostic): (int AS1*, int AS3*, imm, imm)
typedef __attribute__((address_space(1))) int async_gint;
typedef __attribute__((address_space(3))) int async_lint;
#else
#define USE_ASYNC_LDS 0
#endif

// ---------------------------------------------------------------------------
// Kernel 1: spatial mean pool  q[B,C,T,H,W] -> qg stored as qg[c*128 + (b*T+t)]
// One wave (32 threads) per (b,c,t).
// ---------------------------------------------------------------------------
__global__ void pool_kernel(const float* __restrict__ q, float* __restrict__ qg) {
  const int idx = blockIdx.x;                 // ((b*C + c)*T + t)
  const float* base = q + (size_t)idx * HW;
  const int lane = threadIdx.x;               // 32 threads
  float s = 0.0f;
  for (int i = lane; i < HW; i += 32) s += base[i];
  #pragma unroll
  for (int off = 16; off > 0; off >>= 1) s += __shfl_xor(s, off, 32);
  if (lane == 0) {
    const int t = idx % T;
    const int bc = idx / T;
    const int c = bc % C;
    const int b = bc / C;
    qg[c * N + b * T + t] = s * (1.0f / (float)HW);
  }
}

// ---------------------------------------------------------------------------
// Kernel 2: h[o][n] = GELU( pre_w[o,:] @ qg[:,n] + pre_b[o] ),  O=K=128, N=128
// WMMA f32 16x16x4: each wave owns one 16x16 output tile, K stepped by 4.
// ---------------------------------------------------------------------------
__global__ void gemm_gelu_kernel(const float* __restrict__ pre_w,
                                 const float* __restrict__ pre_b,
                                 const float* __restrict__ qg,
                                 float* __restrict__ h_out) {
  const int lane = threadIdx.x & 31;
  const int wave = blockIdx.x * (blockDim.x >> 5) + (threadIdx.x >> 5);
  const int tile_o = (wave >> 3) << 4;        // 8 tiles along O
  const int tile_n = (wave & 7) << 4;         // 8 tiles along N
  const int half = lane >> 4;                 // 0: lanes0-15, 1: lanes16-31
  const int l16 = lane & 15;

#if __has_builtin(__builtin_amdgcn_wmma_f32_16x16x4_f32)
  v8f acc = {};
  for (int kk = 0; kk < C; kk += 4) {
    const int ka = kk + (half ? 2 : 0);
    v2f a, b;
    a.x = pre_w[(tile_o + l16) * C + ka];       // A[M, ka]
    a.y = pre_w[(tile_o + l16) * C + ka + 1];   // A[M, ka+1]
    b.x = qg[(size_t)ka * N + tile_n + l16];    // B[ka,   n]
    b.y = qg[(size_t)(ka + 1) * N + tile_n + l16];
    acc = __builtin_amdgcn_wmma_f32_16x16x4_f32(
        /*neg_a=*/false, a, /*neg_b=*/false, b,
        /*c_mod=*/(short)0, acc, /*reuse_a=*/false, /*reuse_b=*/false);
  }
  #pragma unroll
  for (int r = 0; r < 8; ++r) {
    const int o = tile_o + r + half * 8;
    const int n = tile_n + l16;
    float x = acc[r] + pre_b[o];
    float g = 0.5f * x * (1.0f + erff(x * 0.70710678118654752f)); // exact GELU
    h_out[o * N + n] = g;
  }
#else
  for (int e = lane; e < 256; e += 32) {      // scalar fallback
    const int r = e >> 4, cN = e & 15;
    const int o = tile_o + r, n = tile_n + cN;
    float x = pre_b[o];
    for (int k2 = 0; k2 < C; ++k2) x = fmaf(pre_w[o * C + k2], qg[(size_t)k2 * N + n], x);
    h_out[o * N + n] = 0.5f * x * (1.0f + erff(x * 0.70710678118654752f));
  }
#endif
}

// ---------------------------------------------------------------------------
// Kernel 3: logits[b,m,t] = sum_c sum_j mix_w[m,c,j]*h[c, b*T + t-2+j] + mix_b
// then softmax over m. One block of 128 threads per (b,t).
// ---------------------------------------------------------------------------
__global__ void mix_softmax_kernel(const float* __restrict__ h,
                                   const float* __restrict__ mix_w,
                                   const float* __restrict__ mix_b,
                                   float* __restrict__ alpha) {
  const int n = blockIdx.x;                   // b*T + t
  const int b = n / T, t = n % T;
  const int c = threadIdx.x;                  // 128 threads, one channel each
  float p0 = 0.f, p1 = 0.f, p2 = 0.f;
  #pragma unroll
  for (int j = 0; j < MIXK; ++j) {
    const int tt = t - (MIXK - 1) + j;        // causal padding (2,0)
    if (tt < 0) continue;
    const float hv = h[c * N + b * T + tt];
    p0 = fmaf(mix_w[(0 * C + c) * MIXK + j], hv, p0);
    p1 = fmaf(mix_w[(1 * C + c) * MIXK + j], hv, p1);
    p2 = fmaf(mix_w[(2 * C + c) * MIXK + j], hv, p2);
  }
  __shared__ float red[3][128];
  red[0][c] = p0; red[1][c] = p1; red[2][c] = p2;
  __syncthreads();
  for (int s = 64; s > 0; s >>= 1) {
    if (c < s) {
      red[0][c] += red[0][c + s];
      red[1][c] += red[1][c + s];
      red[2][c] += red[2][c + s];
    }
    __syncthreads();
  }
  if (c == 0) {
    const float l0 = red[0][0] + mix_b[0];
    const float l1 = red[1][0] + mix_b[1];
    const float l2 = red[2][0] + mix_b[2];
    const float mx = fmaxf(l0, fmaxf(l1, l2));
    const float e0 = expf(l0 - mx), e1 = expf(l1 - mx), e2 = expf(l2 - mx);
    const float inv = 1.0f / (e0 + e1 + e2);
    alpha[(b * M + 0) * T + t] = e0 * inv;
    alpha[(b * M + 1) * T + t] = e1 * inv;
    alpha[(b * M + 2) * T + t] = e2 * inv;
  }
}

// ---------------------------------------------------------------------------
// Kernel 4: fused premixed depthwise conv, t-chunked with a 3-slot LDS ring.
// One block (256 thr) per (b, c, chunk-of-8-timesteps). Each input plane is
// read ~1.25x instead of 3x. Plane staging uses GLOBAL_LOAD_ASYNC_TO_LDS_B32
// (ASYNCcnt + s_wait_asynccnt) when available.
// ---------------------------------------------------------------------------
__global__ void conv_mix_kernel(const float* __restrict__ k_in,
                                const float* __restrict__ v_in,
                                const float* __restrict__ Wk,
                                const float* __restrict__ Wv,
                                const float* __restrict__ alpha,
                                float* __restrict__ out) {
  const int blk = blockIdx.x;                 // ((b*C + c)*NCHUNK + chunk)
  const int chunk = blk % NCHUNK;
  const int bc = blk / NCHUNK;
  const int c = bc % C;
  const int b = bc / C;
  const int t0 = chunk * CHUNK;

  __shared__ float sk[KT][PW * PW];           // ring slots, k
  __shared__ float sv[KT][PW * PW];           // ring slots, v
  __shared__ float wke[27];
  __shared__ float wve[27];

  const int tid = threadIdx.x;                // 256 threads
  const size_t pb = ((size_t)(b * C + c)) * T;

  // Zero all ring slots once: borders stay 0 forever (spatial zero-pad), and
  // never-staged slots (t' < 0) implement the causal zero-pad.
  for (int e = tid; e < KT * PW * PW; e += 256) {
    sk[e / (PW * PW)][e % (PW * PW)] = 0.f;
    sv[e / (PW * PW)][e % (PW * PW)] = 0.f;
  }
  __syncthreads();

  for (int i = 0; i < CHUNK + KT - 1; ++i) {  // 10 pipeline steps
    const int tload = t0 - (KT - 1) + i;
    // ---- stage plane `tload` into ring slot (interior 24x24 only) ----
    if (tload >= 0) {
      const int slot = tload % 3;
      const float* gk = k_in + (pb + tload) * HW;
      const float* gv = v_in + (pb + tload) * HW;
      for (int e = tid; e < HW; e += 256) {
        const int li = (e / W + 1) * PW + (e % W + 1);
#if USE_ASYNC_LDS
        __builtin_amdgcn_global_load_async_to_lds_b32(
            (async_gint*)(gk + e), (async_lint*)&sk[slot][li],
            /*offset=*/0, /*cpol=*/0);
        __builtin_amdgcn_global_load_async_to_lds_b32(
            (async_gint*)(gv + e), (async_lint*)&sv[slot][li],
            /*offset=*/0, /*cpol=*/0);
#else
        sk[slot][li] = gk[e];
        sv[slot][li] = gv[e];
#endif
      }
    }

    const bool compute = (i >= KT - 1);
    const int t = t0 + i - (KT - 1);

    // ---- premix the 3 weight banks with alpha[b,:,t] (conv linear in W) ----
    if (compute && tid < 54) {
      const int tap = tid % 27;
      const float* Wsrc = (tid < 27) ? Wk : Wv;
      float wsum = 0.f;
      #pragma unroll
      for (int m = 0; m < M; ++m)
        wsum = fmaf(alpha[(b * M + m) * T + t],
                    Wsrc[((size_t)m * C + c) * 27 + tap], wsum);
      if (tid < 27) wke[tap] = wsum; else wve[tap] = wsum;
    }

#if USE_ASYNC_LDS
    asm volatile("s_wait_asynccnt 0x0" ::: "memory");
#endif
    __syncthreads();                          // plane + weights visible

    // ---- 27-tap conv for output time t ----
    if (compute) {
      #pragma unroll
      for (int o = tid; o < HW; o += 256) {
        const int oh = o / W, ow = o % W;
        float ak = 0.f, av = 0.f;
        #pragma unroll
        for (int p = 0; p < KT; ++p) {
          const int slot = ((t - (KT - 1) + p) % 3 + 3) % 3;
          #pragma unroll
          for (int dh = 0; dh < KS; ++dh)
            #pragma unroll
            for (int dw = 0; dw < KS; ++dw) {
              const int tap = p * 9 + dh * 3 + dw;
              const int si = (oh + dh) * PW + (ow + dw);
              ak = fmaf(wke[tap], sk[slot][si], ak);
              av = fmaf(wve[tap], sv[slot][si], av);
            }
        }
        const size_t g = (pb + t) * HW + o;
        out[g] = ak;                          // k_out
        out[PLANE_ELEMS + g] = av;            // v_out
      }
    }
    __syncthreads();                          // reads done before slot reuse
  }
}

// ---------------------------------------------------------------------------
// Host launcher
// ---------------------------------------------------------------------------
extern "C" void kernel_launch(void* const* d_in, const int* in_sizes, int n_in,
                              void* d_out, int out_size, void* d_ws, size_t ws_size,
                              hipStream_t stream) {
  const float* q     = (const float*)d_in[0];
  const float* k     = (const float*)d_in[1];
  const float* v     = (const float*)d_in[2];
  const float* Wk    = (const float*)d_in[3];
  const float* Wv    = (const float*)d_in[4];
  const float* pre_w = (const float*)d_in[5];
  const float* pre_b = (const float*)d_in[6];
  const float* mix_w = (const float*)d_in[7];
  const float* mix_b = (const float*)d_in[8];
  float* out = (float*)d_out;

  // Workspace layout (floats): qg[0..16384), h[16384..32768), alpha[32768..33152)
  float* ws    = (float*)d_ws;
  float* qg    = ws;
  float* h     = ws + (size_t)C * N;          // 16384
  float* alpha = ws + 2 * (size_t)C * N;      // 32768

  pool_kernel<<<B * C * T, 32, 0, stream>>>(q, qg);
  gemm_gelu_kernel<<<16, 128, 0, stream>>>(pre_w, pre_b, qg, h);
  mix_softmax_kernel<<<B * T, C, 0, stream>>>(h, mix_w, mix_b, alpha);
  conv_mix_kernel<<<B * C * NCHUNK, 256, 0, stream>>>(k, v, Wk, Wv, alpha, out);
}